// DPLRSSMBlock_41214506172567
// MI455X (gfx1250) — compile-verified
//
#include <hip/hip_runtime.h>

// ---------------------------------------------------------------------------
// DPLR SSM block for MI455X (gfx1250, wave32).
//   A = diag(a) + P Q^T  (H x H, rank-4 low-rank part)
//   bx[b,t,:] = B @ x[b,t,:]          (big GEMM, f32 WMMA 16x16x4, TDM-staged)
//   h_t = A h_{t-1} + bx_t            (chunked scan, DPLR fast step)
//
// Pipeline (all on `stream`, graph-capture safe):
//   1) gemm_bx_wmma       : d_out <- x @ Bmat^T   (TDM async tiles -> LDS,
//                           double buffered, f32 WMMA compute)
//   2) compute_AT         : ws.buf0 <- A^T
//   3) 7x gemm_f32_wmma   : (A^T)^128 by repeated squaring -> ALT
//   4) scan_local         : per-chunk local scans (zero init), in place in d_out,
//                           chunk-final states -> ws.hlocEnd
//   5) scan_carry         : serial scan over 32 chunk carries using f32 WMMA
//                           (16x512 state in LDS) -> ws.carries
//   6) scan_correct       : out[b,t,:] += A^{t-t0+1} * carry[chunk]  (DPLR steps)
// ---------------------------------------------------------------------------

#define H_DIM   512
#define R_DIM   4
#define B_DIM   16
#define T_DIM   4096
#define CHUNK   128                  // 2^7 -> 7 squarings build A^128
#define NCHUNK  (T_DIM / CHUNK)      // 32

typedef float v2f __attribute__((ext_vector_type(2)));
typedef float v8f __attribute__((ext_vector_type(8)));
typedef unsigned int u32x4 __attribute__((ext_vector_type(4)));
typedef int i32x4 __attribute__((ext_vector_type(4)));
typedef int i32x8 __attribute__((ext_vector_type(8)));

#define BM 128
#define BN 64
#define BK 32

#if __has_builtin(__builtin_amdgcn_tensor_load_to_lds) && \
    __has_builtin(__builtin_amdgcn_s_wait_tensorcnt)
#define HAS_TDM 1
#else
#define HAS_TDM 0
#endif

#if HAS_TDM
// Issue one 2D TDM tile load: tile_h rows x tile_w f32 elements, row stride
// row_stride_elems in global memory, destination LDS rows padded to tile_w+1
// dwords (pad_interval = 32 dwords, pad_amount = 1 dword).
// D# layout per CDNA5 ISA.
// This toolchain exposes the 6-arg builtin form:
//   (uint32x4 g0, int32x8 g1, int32x4 g2, int32x4 g3, int32x8 gx, i32 cpol)
__device__ __forceinline__ void tdm_load_tile_2d(unsigned lds_addr,
                                                 const void* gptr,
                                                 unsigned tile_w,
                                                 unsigned tile_h,
                                                 unsigned row_stride_elems) {
  unsigned long long ga = (unsigned long long)(size_t)gptr;
  u32x4 g0;
  g0.x = 1u;                                     // count=1, user descriptor
  g0.y = lds_addr;                               // LDS byte address
  g0.z = (unsigned)(ga & 0xffffffffull);         // global_addr lo
  g0.w = (unsigned)(ga >> 32) | (2u << 30);      // global_addr hi | type=2
  const unsigned TD0 = 1u << 20;                 // generous dims: no OOB clip
  const unsigned TD1 = 1u << 20;
  i32x8 g1;
  // [15:0] wg_mask=0 | [17:16] data_size=2 (4B) | [20] pad_enable=1 |
  // [24:22] pad_interval=4 (32 dwords) | [31:25] pad_amount=0 (1 dword)
  g1[0] = (int)((2u << 16) | (1u << 20) | (4u << 22));
  g1[1] = (int)((TD0 & 0xffffu) << 16);          // tensor_dim0 lo16 (bits 79:48)
  g1[2] = (int)(((TD0 >> 16) & 0xffffu) | ((TD1 & 0xffffu) << 16));
  g1[3] = (int)(((TD1 >> 16) & 0xffffu) | (tile_w << 16));   // tile_dim0
  g1[4] = (int)(tile_h & 0xffffu);               // tile_dim1; tile_dim2=0
  g1[5] = (int)row_stride_elems;                 // tensor_dim0_stride lo32
  g1[6] = 0;
  g1[7] = 0;
  i32x4 z4 = {0, 0, 0, 0};                       // groups 2/3 unused (2D tile)
  i32x8 z8 = {0, 0, 0, 0, 0, 0, 0, 0};
  __builtin_amdgcn_tensor_load_to_lds(g0, g1, z4, z4, z8, 0);
}
#endif

// ------------------------- bx GEMM (TDM + WMMA) ----------------------------
// C[M,N] = X[M,K] * Y[N,K]^T with M=B*T, N=K=H. Fixed leading dims = H.
// Double-buffered LDS tiles fetched by the Tensor Data Mover (wave0: X tile,
// wave1: Y tile), tracked with TENSORcnt; all 8 waves compute f32 WMMA.
__global__ __launch_bounds__(256)
void gemm_bx_wmma(const float* __restrict__ X, const float* __restrict__ Y,
                  float* __restrict__ C) {
  __shared__ float Xs[2][BM][BK + 1];            // 2 x 128 x 33 f32
  __shared__ float Ys[2][BN][BK + 1];            // 2 x  64 x 33 f32 (untransposed)

  const int tid  = threadIdx.x;
  const int wave = tid >> 5;
  const int lane = tid & 31;
  const int half = lane >> 4;
  const int lm   = lane & 15;

  const int m0 = blockIdx.y * BM;
  const int n0 = blockIdx.x * BN;
  const int NK = H_DIM / BK;                     // 16 k-tiles

  v8f acc[4] = {};

#if HAS_TDM
  // prologue: fetch tile 0 into buffer 0
  if (wave == 0)
    tdm_load_tile_2d((unsigned)(size_t)&Xs[0][0][0],
                     X + (size_t)m0 * H_DIM, BK, BM, H_DIM);
  else if (wave == 1)
    tdm_load_tile_2d((unsigned)(size_t)&Ys[0][0][0],
                     Y + (size_t)n0 * H_DIM, BK, BN, H_DIM);
#endif

  for (int it = 0; it < NK; ++it) {
    const int buf = it & 1;
#if HAS_TDM
    if (wave <= 1) {
      if (it + 1 < NK) {                         // prefetch tile it+1
        const int k0n = (it + 1) * BK;
        if (wave == 0)
          tdm_load_tile_2d((unsigned)(size_t)&Xs[buf ^ 1][0][0],
                           X + (size_t)m0 * H_DIM + k0n, BK, BM, H_DIM);
        else
          tdm_load_tile_2d((unsigned)(size_t)&Ys[buf ^ 1][0][0],
                           Y + (size_t)n0 * H_DIM + k0n, BK, BN, H_DIM);
        __builtin_amdgcn_s_wait_tensorcnt((short)1);  // tile `it` landed
      } else {
        __builtin_amdgcn_s_wait_tensorcnt((short)0);
      }
    }
    __syncthreads();                             // LDS tile visible to all waves
#else
    // Fallback staging through VGPRs (no TDM builtin on this toolchain)
    const int k0 = it * BK;
#pragma unroll
    for (int i = 0; i < 4; ++i) {
      int linear = tid + i * 256;
      int row = linear >> 3;
      int c4  = (linear & 7) << 2;
      float4 v = *reinterpret_cast<const float4*>(
          &X[(size_t)(m0 + row) * H_DIM + k0 + c4]);
      Xs[buf][row][c4 + 0] = v.x; Xs[buf][row][c4 + 1] = v.y;
      Xs[buf][row][c4 + 2] = v.z; Xs[buf][row][c4 + 3] = v.w;
    }
#pragma unroll
    for (int i = 0; i < 2; ++i) {
      int linear = tid + i * 256;
      int n  = linear >> 3;
      int c4 = (linear & 7) << 2;
      float4 v = *reinterpret_cast<const float4*>(
          &Y[(size_t)(n0 + n) * H_DIM + k0 + c4]);
      Ys[buf][n][c4 + 0] = v.x; Ys[buf][n][c4 + 1] = v.y;
      Ys[buf][n][c4 + 2] = v.z; Ys[buf][n][c4 + 3] = v.w;
    }
    __syncthreads();
#endif

    const int mrow = wave * 16 + lm;
#pragma unroll
    for (int kk = 0; kk < BK; kk += 4) {
      v2f a;                                     // A 16x4: lanes0-15 K=0,1; 16-31 K=2,3
      a.x = Xs[buf][mrow][kk + 2 * half + 0];
      a.y = Xs[buf][mrow][kk + 2 * half + 1];
#pragma unroll
      for (int nt = 0; nt < 4; ++nt) {           // B[k][n] = Y[n][k] (columns of Ys)
        v2f b;
        b.x = Ys[buf][nt * 16 + lm][kk + 2 * half + 0];
        b.y = Ys[buf][nt * 16 + lm][kk + 2 * half + 1];
        acc[nt] = __builtin_amdgcn_wmma_f32_16x16x4_f32(
            false, a, false, b, (short)0, acc[nt], false, false);
      }
    }
    __syncthreads();                             // protect buffer for next TDM write
  }

#pragma unroll
  for (int nt = 0; nt < 4; ++nt) {
#pragma unroll
    for (int j = 0; j < 8; ++j) {
      int row = m0 + wave * 16 + half * 8 + j;   // C/D: vgpr j -> M=j (+8 upper half)
      int col = n0 + nt * 16 + lm;
      C[(size_t)row * H_DIM + col] = acc[nt][j];
    }
  }
}

// ------------------------- generic WMMA GEMM (NN) --------------------------
// C[M,N] = X[M,K] * Y[K,N] — used for the 7 squarings of A^T (512^3 each).
__global__ __launch_bounds__(256)
void gemm_f32_wmma(const float* __restrict__ X, const float* __restrict__ Y,
                   float* __restrict__ C, int M, int N, int K,
                   int ldx, int ldy, int ldc) {
  __shared__ float Xs[BM][BK + 1];
  __shared__ float Ys[BK][BN + 1];

  const int tid  = threadIdx.x;
  const int wave = tid >> 5;
  const int lane = tid & 31;
  const int half = lane >> 4;
  const int lm   = lane & 15;

  const int m0 = blockIdx.y * BM;
  const int n0 = blockIdx.x * BN;

  v8f acc[4] = {};

  for (int k0 = 0; k0 < K; k0 += BK) {
#pragma unroll
    for (int i = 0; i < 4; ++i) {
      int linear = tid + i * 256;
      int row    = linear >> 3;
      int c4     = (linear & 7) << 2;
      float4 v = *reinterpret_cast<const float4*>(
          &X[(size_t)(m0 + row) * ldx + k0 + c4]);
      Xs[row][c4 + 0] = v.x; Xs[row][c4 + 1] = v.y;
      Xs[row][c4 + 2] = v.z; Xs[row][c4 + 3] = v.w;
    }
#pragma unroll
    for (int i = 0; i < 2; ++i) {
      int linear = tid + i * 256;
      int k      = linear >> 4;
      int c4     = (linear & 15) << 2;
      float4 v = *reinterpret_cast<const float4*>(
          &Y[(size_t)(k0 + k) * ldy + n0 + c4]);
      Ys[k][c4 + 0] = v.x; Ys[k][c4 + 1] = v.y;
      Ys[k][c4 + 2] = v.z; Ys[k][c4 + 3] = v.w;
    }
    __syncthreads();

    const int mrow = wave * 16 + lm;
#pragma unroll
    for (int kk = 0; kk < BK; kk += 4) {
      v2f a;
      a.x = Xs[mrow][kk + 2 * half + 0];
      a.y = Xs[mrow][kk + 2 * half + 1];
#pragma unroll
      for (int nt = 0; nt < 4; ++nt) {
        v2f b;
        b.x = Ys[kk + 2 * half + 0][nt * 16 + lm];
        b.y = Ys[kk + 2 * half + 1][nt * 16 + lm];
        acc[nt] = __builtin_amdgcn_wmma_f32_16x16x4_f32(
            false, a, false, b, (short)0, acc[nt], false, false);
      }
    }
    __syncthreads();
  }

#pragma unroll
  for (int nt = 0; nt < 4; ++nt) {
#pragma unroll
    for (int j = 0; j < 8; ++j) {
      int row = m0 + wave * 16 + half * 8 + j;
      int col = n0 + nt * 16 + lm;
      C[(size_t)row * ldc + col] = acc[nt][j];
    }
  }
}

// --------------------------- A^T materialization ---------------------------
__global__ __launch_bounds__(256)
void compute_AT(const float* __restrict__ a, const float* __restrict__ P,
                const float* __restrict__ Q, float* __restrict__ AT) {
  int idx = blockIdx.x * 256 + threadIdx.x;
  int r = idx >> 9;                              // AT row = A column j
  int c = idx & (H_DIM - 1);                     // AT col = A row i
  float4 p = *reinterpret_cast<const float4*>(&P[c * R_DIM]);
  float4 q = *reinterpret_cast<const float4*>(&Q[r * R_DIM]);
  float v = p.x * q.x + p.y * q.y + p.z * q.z + p.w * q.w;
  if (c == r) v += a[c];
  AT[idx] = v;
}

// ------------------------------ local scans --------------------------------
__global__ __launch_bounds__(512)
void scan_local(float* __restrict__ out, const float* __restrict__ a_diag,
                const float* __restrict__ P, const float* __restrict__ Q,
                float* __restrict__ hlocEnd) {
  const int c    = blockIdx.x;
  const int b    = threadIdx.x >> 5;
  const int lane = threadIdx.x & 31;
  const int cb   = lane * 16;

  float av[16], h[16];
  float4 Pr[16], Qr[16];
#pragma unroll
  for (int q = 0; q < 4; ++q) {
    float4 t = *reinterpret_cast<const float4*>(&a_diag[cb + 4 * q]);
    av[4 * q + 0] = t.x; av[4 * q + 1] = t.y; av[4 * q + 2] = t.z; av[4 * q + 3] = t.w;
  }
#pragma unroll
  for (int j = 0; j < 16; ++j) {
    Pr[j] = *reinterpret_cast<const float4*>(&P[(cb + j) * R_DIM]);
    Qr[j] = *reinterpret_cast<const float4*>(&Q[(cb + j) * R_DIM]);
    h[j] = 0.0f;
  }

  size_t base = ((size_t)b * T_DIM + (size_t)c * CHUNK) * H_DIM + cb;
  for (int s = 0; s < CHUNK; ++s) {
    float* row = out + base + (size_t)s * H_DIM;
    if (s + 1 < CHUNK) __builtin_prefetch(row + H_DIM, 0, 1);  // next bx row
    float bxv[16];
#pragma unroll
    for (int q = 0; q < 4; ++q) {
      float4 t = *reinterpret_cast<const float4*>(row + 4 * q);
      bxv[4 * q + 0] = t.x; bxv[4 * q + 1] = t.y;
      bxv[4 * q + 2] = t.z; bxv[4 * q + 3] = t.w;
    }
    float ux = 0.f, uy = 0.f, uz = 0.f, uw = 0.f;
#pragma unroll
    for (int j = 0; j < 16; ++j) {
      ux = fmaf(h[j], Qr[j].x, ux); uy = fmaf(h[j], Qr[j].y, uy);
      uz = fmaf(h[j], Qr[j].z, uz); uw = fmaf(h[j], Qr[j].w, uw);
    }
#pragma unroll
    for (int off = 16; off >= 1; off >>= 1) {
      ux += __shfl_xor(ux, off, 32); uy += __shfl_xor(uy, off, 32);
      uz += __shfl_xor(uz, off, 32); uw += __shfl_xor(uw, off, 32);
    }
#pragma unroll
    for (int j = 0; j < 16; ++j) {
      float v = fmaf(av[j], h[j], bxv[j]);
      v = fmaf(Pr[j].x, ux, v); v = fmaf(Pr[j].y, uy, v);
      v = fmaf(Pr[j].z, uz, v); v = fmaf(Pr[j].w, uw, v);
      h[j] = v;
    }
#pragma unroll
    for (int q = 0; q < 4; ++q) {
      float4 t = make_float4(h[4*q+0], h[4*q+1], h[4*q+2], h[4*q+3]);
      *reinterpret_cast<float4*>(row + 4 * q) = t;
    }
  }
  float* he = hlocEnd + ((size_t)c * B_DIM + b) * H_DIM + cb;
#pragma unroll
  for (int q = 0; q < 4; ++q) {
    float4 t = make_float4(h[4*q+0], h[4*q+1], h[4*q+2], h[4*q+3]);
    *reinterpret_cast<float4*>(he + 4 * q) = t;
  }
}

// ------------------------------ carry scan ---------------------------------
__global__ __launch_bounds__(512)
void scan_carry(const float* __restrict__ ALT, const float* __restrict__ hlocEnd,
                float* __restrict__ carries) {
  __shared__ float cs[2][B_DIM][H_DIM + 4];      // +4 pad: stride 516, no conflicts

  const int tid  = threadIdx.x;
  const int wave = tid >> 5;
  const int lane = tid & 31;
  const int half = lane >> 4;
  const int lm   = lane & 15;

  float* flat0 = &cs[0][0][0];
  for (int i = tid; i < B_DIM * (H_DIM + 4); i += 512) flat0[i] = 0.0f;
  __syncthreads();

  int cur = 0;
  for (int c = 0; c < NCHUNK; ++c) {
#pragma unroll
    for (int j = 0; j < 16; ++j) {               // emit carry (state before chunk c)
      int i = lane + j * 32;
      carries[((size_t)c * B_DIM + wave) * H_DIM + i] = cs[cur][wave][i];
    }

    v8f acc[2] = {};
#pragma unroll
    for (int nt = 0; nt < 2; ++nt) {
      int n0 = wave * 32 + nt * 16;
#pragma unroll
      for (int j = 0; j < 8; ++j) {              // preload C with hlocEnd
        int m = j + 8 * half;
        acc[nt][j] = hlocEnd[((size_t)c * B_DIM + m) * H_DIM + n0 + lm];
      }
    }
#pragma unroll 4
    for (int k = 0; k < H_DIM; k += 4) {
      v2f a;
      a.x = cs[cur][lm][k + 2 * half + 0];
      a.y = cs[cur][lm][k + 2 * half + 1];
#pragma unroll
      for (int nt = 0; nt < 2; ++nt) {
        int n0 = wave * 32 + nt * 16;
        v2f bb;
        bb.x = ALT[(size_t)(k + 2 * half + 0) * H_DIM + n0 + lm];
        bb.y = ALT[(size_t)(k + 2 * half + 1) * H_DIM + n0 + lm];
        acc[nt] = __builtin_amdgcn_wmma_f32_16x16x4_f32(
            false, a, false, bb, (short)0, acc[nt], false, false);
      }
    }
    __syncthreads();
#pragma unroll
    for (int nt = 0; nt < 2; ++nt) {
      int n0 = wave * 32 + nt * 16;
#pragma unroll
      for (int j = 0; j < 8; ++j) {
        int m = j + 8 * half;
        cs[1 - cur][m][n0 + lm] = acc[nt][j];
      }
    }
    __syncthreads();
    cur ^= 1;
  }
}

// ----------------------------- correction pass -----------------------------
__global__ __launch_bounds__(512)
void scan_correct(float* __restrict__ out, const float* __restrict__ a_diag,
                  const float* __restrict__ P, const float* __restrict__ Q,
                  const float* __restrict__ carries) {
  const int c = blockIdx.x;
  if (c == 0) return;                            // carry[0] == 0
  const int b    = threadIdx.x >> 5;
  const int lane = threadIdx.x & 31;
  const int cb   = lane * 16;

  float av[16], v[16];
  float4 Pr[16], Qr[16];
#pragma unroll
  for (int q = 0; q < 4; ++q) {
    float4 t = *reinterpret_cast<const float4*>(&a_diag[cb + 4 * q]);
    av[4 * q + 0] = t.x; av[4 * q + 1] = t.y; av[4 * q + 2] = t.z; av[4 * q + 3] = t.w;
  }
#pragma unroll
  for (int j = 0; j < 16; ++j) {
    Pr[j] = *reinterpret_cast<const float4*>(&P[(cb + j) * R_DIM]);
    Qr[j] = *reinterpret_cast<const float4*>(&Q[(cb + j) * R_DIM]);
  }
  const float* cr = carries + ((size_t)c * B_DIM + b) * H_DIM + cb;
#pragma unroll
  for (int q = 0; q < 4; ++q) {
    float4 t = *reinterpret_cast<const float4*>(cr + 4 * q);
    v[4 * q + 0] = t.x; v[4 * q + 1] = t.y; v[4 * q + 2] = t.z; v[4 * q + 3] = t.w;
  }

  size_t base = ((size_t)b * T_DIM + (size_t)c * CHUNK) * H_DIM + cb;
  for (int s = 0; s < CHUNK; ++s) {
    float* row = out + base + (size_t)s * H_DIM;
    if (s + 1 < CHUNK) __builtin_prefetch(row + H_DIM, 1, 1);
    float ux = 0.f, uy = 0.f, uz = 0.f, uw = 0.f;
#pragma unroll
    for (int j = 0; j < 16; ++j) {
      ux = fmaf(v[j], Qr[j].x, ux); uy = fmaf(v[j], Qr[j].y, uy);
      uz = fmaf(v[j], Qr[j].z, uz); uw = fmaf(v[j], Qr[j].w, uw);
    }
#pragma unroll
    for (int off = 16; off >= 1; off >>= 1) {
      ux += __shfl_xor(ux, off, 32); uy += __shfl_xor(uy, off, 32);
      uz += __shfl_xor(uz, off, 32); uw += __shfl_xor(uw, off, 32);
    }
#pragma unroll
    for (int j = 0; j < 16; ++j) {
      float t = av[j] * v[j];
      t = fmaf(Pr[j].x, ux, t); t = fmaf(Pr[j].y, uy, t);
      t = fmaf(Pr[j].z, uz, t); t = fmaf(Pr[j].w, uw, t);
      v[j] = t;
    }
#pragma unroll
    for (int q = 0; q < 4; ++q) {
      float4 t = *reinterpret_cast<const float4*>(row + 4 * q);
      t.x += v[4*q+0]; t.y += v[4*q+1]; t.z += v[4*q+2]; t.w += v[4*q+3];
      *reinterpret_cast<float4*>(row + 4 * q) = t;
    }
  }
}

// ------------------------------ launch glue --------------------------------
extern "C" void kernel_launch(void* const* d_in, const int* in_sizes, int n_in,
                              void* d_out, int out_size, void* d_ws, size_t ws_size,
                              hipStream_t stream) {
  const float* x  = (const float*)d_in[0];   // [B,T,H]
  const float* a  = (const float*)d_in[1];   // [H]
  const float* P  = (const float*)d_in[2];   // [H,R]
  const float* Q  = (const float*)d_in[3];   // [H,R]
  const float* Bm = (const float*)d_in[4];   // [H,H]
  float* out = (float*)d_out;                // [B,T,H] — also serves as bx buffer

  const size_t MAT = (size_t)H_DIM * H_DIM;
  float* buf0    = (float*)d_ws;
  float* buf1    = buf0 + MAT;
  float* hlocEnd = buf1 + MAT;                                  // [NC,B,H]
  float* carries = hlocEnd + (size_t)NCHUNK * B_DIM * H_DIM;    // [NC,B,H]
  // total workspace use: 4 MB

  // 1) bx = x @ Bm^T -> out  (TDM-staged WMMA GEMM)
  {
    dim3 grid(H_DIM / BN, (B_DIM * T_DIM) / BM);  // (8, 512)
    gemm_bx_wmma<<<grid, 256, 0, stream>>>(x, Bm, out);
  }
  // 2) A^T
  compute_AT<<<(H_DIM * H_DIM) / 256, 256, 0, stream>>>(a, P, Q, buf0);
  // 3) (A^T)^128 by 7 squarings (ping-pong); result ends in `src` after loop
  float* src = buf0;
  float* dst = buf1;
  for (int i = 0; i < 7; ++i) {
    dim3 grid(H_DIM / BN, H_DIM / BM);            // (8, 4)
    gemm_f32_wmma<<<grid, 256, 0, stream>>>(
        src, src, dst, H_DIM, H_DIM, H_DIM, H_DIM, H_DIM, H_DIM);
    float* t = src; src = dst; dst = t;
  }
  const float* ALT = src;
  // 4) local scans (parallel over 32 chunks x 16 batches)
  scan_local<<<NCHUNK, 512, 0, stream>>>(out, a, P, Q, hlocEnd);
  // 5) serial carry scan (32 dense WMMA steps, single workgroup)
  scan_carry<<<1, 512, 0, stream>>>(ALT, hlocEnd, carries);
  // 6) broadcast carries into each chunk
  scan_correct<<<NCHUNK, 512, 0, stream>>>(out, a, P, Q, carries);
}